// InputAverageModel_34574486733038
// MI455X (gfx1250) — compile-verified
//
#include <hip/hip_runtime.h>

typedef __attribute__((ext_vector_type(2)))  float    v2f;
typedef __attribute__((ext_vector_type(8)))  float    v8f;

#define NULLV (-1.0f)
#define NB    16
#define NT    288
#define NN    4096
#define NH    10
#define NR    16

// ws layout (floats)
#define WS_GMEAN  0
#define WS_COUNTS 1     // 16 floats
#define WS_PSUM   64    // 256 floats
#define WS_PCNT   320   // 256 floats

// ---------------------------------------------------------------------------
// Pass 1: stream 151MB of [B,T,N,F] fp32, per-(b,n) valid sum/count.
// T is split in two halves per thread to double wave count (2048 waves /
// 256 workgroups) for HBM latency hiding. Thread j -> (tHalf, b, node pair),
// one float4 (= global_load_b128, 512B per wave per step) per t.
// Partial sums -> out slices h=0/2, counts -> h=1/3 (d_out reused as scratch).
// Block-reduced global sum/count partials -> ws for the global mean.
// ---------------------------------------------------------------------------
__global__ void k_time_reduce(const float* __restrict__ x,
                              float* __restrict__ out,
                              float* __restrict__ ws) {
    const int tid   = threadIdx.x;
    const int j     = blockIdx.x * blockDim.x + tid;  // 0 .. 65535
    const int pair  = j & 2047;
    const int b     = (j >> 11) & (NB - 1);
    const int tHalf = j >> 15;                        // 0 / 1
    const int n0    = pair * 2;
    const int t0    = tHalf * (NT / 2);

    // float4 linear index: (b*NT + t)*2048 + pair
    const float4* p = (const float4*)x + ((size_t)b * NT + t0) * 2048 + pair;

    float s0 = 0.f, s1 = 0.f, c0 = 0.f, c1 = 0.f;
#pragma unroll 4
    for (int t = 0; t < NT / 2; ++t) {
        float4 v = p[(size_t)t * 2048];
        // v.x = feat0 of node n0, v.z = feat0 of node n0+1 (feat1 discarded;
        // it shares cachelines with feat0 so it costs no extra bandwidth)
        bool g0 = (v.x != NULLV);
        bool g1 = (v.z != NULLV);
        s0 += g0 ? v.x : 0.f;  c0 += g0 ? 1.f : 0.f;
        s1 += g1 ? v.z : 0.f;  c1 += g1 ? 1.f : 0.f;
    }

    float* slice = out + (size_t)b * NH * NN + (size_t)(2 * tHalf) * NN;
    slice[n0]          = s0;
    slice[n0 + 1]      = s1;
    slice[NN + n0]     = c0;
    slice[NN + n0 + 1] = c1;

    __shared__ float ssum[256];
    __shared__ float scnt[256];
    ssum[tid] = s0 + s1;
    scnt[tid] = c0 + c1;
    __syncthreads();
#pragma unroll
    for (int off = 128; off > 0; off >>= 1) {
        if (tid < off) { ssum[tid] += ssum[tid + off]; scnt[tid] += scnt[tid + off]; }
        __syncthreads();
    }
    if (tid == 0) {
        ws[WS_PSUM + blockIdx.x] = ssum[0];
        ws[WS_PCNT + blockIdx.x] = scnt[0];
    }
}

// ---------------------------------------------------------------------------
// Pass 1b: finish global mean (256 block partials); per-region node counts.
// ---------------------------------------------------------------------------
__global__ void k_stats(const int* __restrict__ cid, float* __restrict__ ws) {
    __shared__ float ssum[256];
    __shared__ float scnt[256];
    __shared__ int   hist[NR];
    const int tid = threadIdx.x;
    ssum[tid] = ws[WS_PSUM + tid];
    scnt[tid] = ws[WS_PCNT + tid];
    if (tid < NR) hist[tid] = 0;
    __syncthreads();
#pragma unroll
    for (int off = 128; off > 0; off >>= 1) {
        if (tid < off) { ssum[tid] += ssum[tid + off]; scnt[tid] += scnt[tid + off]; }
        __syncthreads();
    }
    for (int i = tid; i < NN; i += 256) atomicAdd(&hist[cid[i] & (NR - 1)], 1);
    __syncthreads();
    if (tid == 0)  ws[WS_GMEAN] = ssum[0] / fmaxf(scnt[0], 1.0f);
    if (tid < NR)  ws[WS_COUNTS + tid] = (float)hist[tid];
}

// ---------------------------------------------------------------------------
// Pass 2: pred = (sum + (T - cnt)*gmean)/T from the two T-half partials,
// broadcast to all 10 horizon slices.
// ---------------------------------------------------------------------------
__global__ void k_pred(float* __restrict__ out, const float* __restrict__ ws) {
    const int j = blockIdx.x * blockDim.x + threadIdx.x;   // 0 .. 65535
    const int b = j >> 12;
    const int n = j & (NN - 1);
    const float gmean = ws[WS_GMEAN];
    float* slice = out + (size_t)b * NH * NN;
    const float s = slice[n]      + slice[2 * NN + n];
    const float c = slice[NN + n] + slice[3 * NN + n];
    const float pred = (s + ((float)NT - c) * gmean) * (1.0f / (float)NT);
#pragma unroll
    for (int h = 0; h < NH; ++h) slice[h * NN + n] = pred;
}

// ---------------------------------------------------------------------------
// Pass 3: regional mean as 16x4096 (pred) x 4096x16 (one-hot cluster) on the
// matrix pipe: V_WMMA_F32_16X16X4_F32, K=4096 split over 8 waves, LDS
// cross-wave reduce, divide by region counts, tile over 10 horizon slots.
// A layout (f32 16x16x4): lane L -> m=L%16, VGPR0=K[2*(L/16)], VGPR1=K[+1].
// B layout mirrored with n=L%16. C/D: VGPR i -> row i+8*(L/16), col L%16.
// ---------------------------------------------------------------------------
__global__ void k_regional(const float* __restrict__ out_pred,
                           const int* __restrict__ cid,
                           const float* __restrict__ ws,
                           float* __restrict__ out_reg) {
    const int tid  = threadIdx.x;
    const int lane = tid & 31;
    const int wave = tid >> 5;        // 0..7
    const int m    = lane & 15;       // A row (batch) == B column (region)
    const int half = lane >> 4;       // 0/1

    v8f c = {};
    const int k_begin = wave * (NN / 8);
    const int k_end   = k_begin + (NN / 8);

    for (int k0 = k_begin; k0 < k_end; k0 += 4) {
        const int ka = k0 + 2 * half;
        v2f a, bm;
        a.x  = out_pred[(size_t)m * NH * NN + ka];       // pred[m][ka] (h=0 slice)
        a.y  = out_pred[(size_t)m * NH * NN + ka + 1];
        bm.x = (cid[ka]     == m) ? 1.0f : 0.0f;
        bm.y = (cid[ka + 1] == m) ? 1.0f : 0.0f;
        c = __builtin_amdgcn_wmma_f32_16x16x4_f32(false, a, false, bm,
                                                  (short)0, c, false, false);
    }

    __shared__ float red[8][32][8];
#pragma unroll
    for (int i = 0; i < 8; ++i) red[wave][lane][i] = c[i];
    __syncthreads();

    if (wave == 0) {
#pragma unroll
        for (int i = 0; i < 8; ++i) {
            float s = 0.f;
#pragma unroll
            for (int w = 0; w < 8; ++w) s += red[w][lane][i];
            const int   row = i + 8 * half;     // batch
            const int   col = m;                // region
            const float val = s / fmaxf(ws[WS_COUNTS + col], 1.0f);
            for (int h = 0; h < NH; ++h)
                out_reg[(row * NH + h) * NR + col] = val;
        }
    }
}

extern "C" void kernel_launch(void* const* d_in, const int* in_sizes, int n_in,
                              void* d_out, int out_size, void* d_ws, size_t ws_size,
                              hipStream_t stream) {
    (void)in_sizes; (void)n_in; (void)out_size; (void)ws_size;
    const float* x   = (const float*)d_in[0];
    const int*   cid = (const int*)d_in[1];
    float* out = (float*)d_out;
    float* ws  = (float*)d_ws;

    k_time_reduce<<<256, 256, 0, stream>>>(x, out, ws);
    k_stats      <<<1,   256, 0, stream>>>(cid, ws);
    k_pred       <<<256, 256, 0, stream>>>(out, ws);
    k_regional   <<<1,   256, 0, stream>>>(out, cid, ws,
                                           out + (size_t)NB * NH * NN);
}